// OptimizedExpertRouter_40089224741097
// MI455X (gfx1250) — compile-verified
//
#include <hip/hip_runtime.h>
#include <hip/hip_bf16.h>
#include <math.h>

typedef float v2f __attribute__((ext_vector_type(2)));
typedef float v8f __attribute__((ext_vector_type(8)));

#define HIDDEN        4096
#define NUM_EXPERTS   64
#define TOK_PER_BLK   16
#define KCHUNK        64
#define NCHUNKS       (HIDDEN / KCHUNK)   // 64
#define AB_STRIDE     68    // 68*4B = 272B row: 16B aligned, %64 banks = 4 -> conflict-free frags
#define L_STRIDE      66    // logits tile stride: wave halves hit disjoint bank sets

// Async global->LDS copy of 16 bytes (GLOBAL_LOAD_ASYNC_TO_LDS_B128, ASYNCcnt-tracked).
// lds_off = LDS byte address (low 32 bits of flat shared pointer), gsrc = 64-bit global addr.
__device__ __forceinline__ void async_copy_b128(unsigned lds_off, const float* gsrc) {
    asm volatile("global_load_async_to_lds_b128 %0, %1, off"
                 :: "v"(lds_off), "v"(gsrc)
                 : "memory");
}
__device__ __forceinline__ unsigned lds_addr(const void* p) {
    return (unsigned)(unsigned long long)p;   // flat LDS addr: low 32 bits = LDS offset
}

__global__ __launch_bounds__(128) void
moe_router_wmma_f32(const float* __restrict__ hs,     // [16384, 4096]
                    const float* __restrict__ W,      // [64, 4096]
                    float* __restrict__ rweights,     // [16384, 2]
                    int*   __restrict__ sel,          // [16384, 2]
                    float* __restrict__ logits_out)   // [16384, 64]
{
    __shared__ float ldsA[2][TOK_PER_BLK  * AB_STRIDE];   // 2 x 4352 B
    __shared__ float ldsB[2][NUM_EXPERTS * AB_STRIDE];    // 2 x 17408 B
    __shared__ float ldsL[TOK_PER_BLK * L_STRIDE];        // 4224 B

    const int  t    = threadIdx.x;        // 0..127
    const int  wave = t >> 5;             // 0..3 -> expert tile
    const int  lane = t & 31;
    const int  h    = lane >> 4;          // half-wave selects K pair
    const int  n    = lane & 15;          // M for A-frag, N (expert-in-tile) for B-frag
    const long tok0 = (long)blockIdx.x * TOK_PER_BLK;

    // Per-thread cooperative-load coordinates (row of 64 floats = 16 float4, 16 threads/row).
    const int arow0 = t >> 4;             // A: 2 float4/thread -> rows t>>4, (t>>4)+8
    const int ac4   = t & 15;
    const int brow0 = t >> 4;             // B: 8 float4/thread -> rows t>>4 + 8*i
    const int bc4   = t & 15;

    // Stage one K-chunk into LDS buffer `buf` with 10 async b128 copies per thread.
    auto stage = [&](int buf, int k0) {
        #pragma unroll
        for (int i = 0; i < 2; ++i) {                       // A: 16x64 floats
            const int row = arow0 + i * 8;
            async_copy_b128(lds_addr(&ldsA[buf][row * AB_STRIDE + ac4 * 4]),
                            hs + (tok0 + row) * HIDDEN + k0 + ac4 * 4);
        }
        #pragma unroll
        for (int i = 0; i < 8; ++i) {                       // B: 64x64 floats
            const int row = brow0 + i * 8;
            async_copy_b128(lds_addr(&ldsB[buf][row * AB_STRIDE + bc4 * 4]),
                            W + (long)row * HIDDEN + k0 + bc4 * 4);
        }
    };

    v8f acc = {};

    stage(0, 0);                                            // prologue: chunk 0
    for (int c = 0; c < NCHUNKS; ++c) {
        const int cur = c & 1;
        if (c + 1 < NCHUNKS) {
            stage(cur ^ 1, (c + 1) * KCHUNK);               // prefetch next chunk
            asm volatile("s_wait_asynccnt 0xa" ::: "memory");   // chunk c's 10 copies done
        } else {
            asm volatile("s_wait_asynccnt 0x0" ::: "memory");
        }
        __syncthreads();                                    // all waves' data visible

        const float* aRow = &ldsA[cur][n * AB_STRIDE + 2 * h];
        const float* bRow = &ldsB[cur][(wave * 16 + n) * AB_STRIDE + 2 * h];
        #pragma unroll
        for (int s = 0; s < KCHUNK / 4; ++s) {              // 16 x V_WMMA_F32_16X16X4_F32
            v2f a = *(const v2f*)(aRow + s * 4);            // A[M=n][k+4s+2h .. +1]
            v2f b = *(const v2f*)(bRow + s * 4);            // B[k+4s+2h .. +1][N=n]
            acc = __builtin_amdgcn_wmma_f32_16x16x4_f32(
                false, a, false, b, (short)0, acc, false, false);
        }
        __syncthreads();                                    // before buffer reuse
    }

    // ---- epilogue: write logits (global, coalesced) and stash tile in LDS ----
    const int expert = wave * 16 + n;
    #pragma unroll
    for (int v = 0; v < 8; ++v) {
        const int M = v + 8 * h;                            // D layout: vgpr v, half h -> row
        logits_out[(tok0 + M) * NUM_EXPERTS + expert] = acc[v];
        ldsL[M * L_STRIDE + expert] = acc[v];
    }
    __syncthreads();

    // ---- softmax + top-2 per token (16 tokens -> threads 0..15) ----
    if (t < TOK_PER_BLK) {
        const float* row = &ldsL[t * L_STRIDE];
        float v1 = -INFINITY, v2 = -INFINITY;
        int   i1 = 0, i2 = 0;
        #pragma unroll 8
        for (int e = 0; e < NUM_EXPERTS; ++e) {
            const float x = row[e];
            if (x > v1)      { v2 = v1; i2 = i1; v1 = x; i1 = e; }
            else if (x > v2) { v2 = x; i2 = e; }
        }
        float ssum = 0.0f;
        #pragma unroll 8
        for (int e = 0; e < NUM_EXPERTS; ++e)
            ssum += __expf(row[e] - v1);
        const float inv_s = 1.0f / ssum;
        const float p1 = inv_s;                             // exp(v1 - v1) / sum
        const float p2 = __expf(v2 - v1) * inv_s;
        const float norm = 1.0f / (p1 + p2 + 1e-6f);
        const long tok = tok0 + t;
        rweights[tok * 2 + 0] = p1 * norm;
        rweights[tok * 2 + 1] = p2 * norm;
        sel[tok * 2 + 0] = i1;
        sel[tok * 2 + 1] = i2;
    }
}

extern "C" void kernel_launch(void* const* d_in, const int* in_sizes, int n_in,
                              void* d_out, int out_size, void* d_ws, size_t ws_size,
                              hipStream_t stream) {
    const float* hs = (const float*)d_in[0];      // [4,4096,4096] f32
    const float* W  = (const float*)d_in[1];      // [64,4096] f32
    const int tokens = in_sizes[0] / HIDDEN;      // 16384

    float* out      = (float*)d_out;
    float* rweights = out;                        // [tokens,2] f32
    int*   sel      = (int*)(out + tokens * 2);   // [tokens,2] i32
    float* logits   = out + tokens * 4;           // [tokens,64] f32

    moe_router_wmma_f32<<<tokens / TOK_PER_BLK, 128, 0, stream>>>(
        hs, W, rweights, sel, logits);
}